// PtsManipulator_60945585931016
// MI455X (gfx1250) — compile-verified
//
#include <hip/hip_runtime.h>
#include <math.h>

#define BATCH 4
#define CCH   64
#define WRES  64
#define NPIX  (WRES*WRES)      // 4096
#define NPTS  (2*NPIX)         // 8192
#define KNN   4
#define EPSV  0.01f

typedef float v2f __attribute__((ext_vector_type(2)));
typedef float v8f __attribute__((ext_vector_type(8)));
typedef unsigned int u32x4 __attribute__((ext_vector_type(4)));
typedef int i32x8 __attribute__((ext_vector_type(8)));
typedef int i32x4 __attribute__((ext_vector_type(4)));

// ---------------------------------------------------------------------------
// Kernel 0: zero output, init z-buffer to +inf
// ---------------------------------------------------------------------------
__global__ void init_kernel(float* out, int out_n, unsigned* zbuf) {
    int i = blockIdx.x * blockDim.x + threadIdx.x;
    if (i < out_n) out[i] = 0.0f;
    if (i < BATCH * NPIX) zbuf[i] = 0x7F800000u;   // +inf
}

// ---------------------------------------------------------------------------
// Kernel 1: unproject both images to world points; pack (x,y,z,|p|^2)
// ---------------------------------------------------------------------------
__global__ void prep_points(const float* __restrict__ pp1, const float* __restrict__ pp2,
                            const float* __restrict__ Kinv,
                            const float* __restrict__ Ri1, const float* __restrict__ Ri2,
                            float4* __restrict__ P) {
    int t = blockIdx.x * blockDim.x + threadIdx.x;
    if (t >= BATCH * NPTS) return;
    int b   = t / NPTS;
    int r   = t - b * NPTS;
    int img = r >> 12;           // 0 or 1
    int pix = r & (NPIX - 1);
    float d = img ? pp2[b * NPIX + pix] : pp1[b * NPIX + pix];
    float fx = (float)(pix & (WRES - 1));
    float fy = (float)(pix >> 6);
    float pr0 = fx * d, pr1 = fy * d, pr2 = d, pr3 = 1.0f;
    const float* Ki = Kinv + b * 16;
    float tv[4];
#pragma unroll
    for (int i = 0; i < 4; ++i)
        tv[i] = Ki[i*4+0]*pr0 + Ki[i*4+1]*pr1 + Ki[i*4+2]*pr2 + Ki[i*4+3]*pr3;
    const float* R = (img ? Ri2 : Ri1) + b * 16;
    float w0 = R[0]*tv[0] + R[1]*tv[1] + R[2] *tv[2] + R[3] *tv[3];
    float w1 = R[4]*tv[0] + R[5]*tv[1] + R[6] *tv[2] + R[7] *tv[3];
    float w2 = R[8]*tv[0] + R[9]*tv[1] + R[10]*tv[2] + R[11]*tv[3];
    P[t] = make_float4(w0, w1, w2, w0*w0 + w1*w1 + w2*w2);
}

// ---------------------------------------------------------------------------
// Kernel 2: build query operands q_m = pts[2m], pre-scaled (-2x,-2y,-2z,1)
// ---------------------------------------------------------------------------
__global__ void prep_queries(const float4* __restrict__ P, float4* __restrict__ Q) {
    int t = blockIdx.x * blockDim.x + threadIdx.x;
    if (t >= BATCH * NPIX) return;
    int b = t >> 12;
    int m = t & (NPIX - 1);
    float4 p = P[b * NPTS + 2 * m];
    Q[t] = make_float4(-2.0f * p.x, -2.0f * p.y, -2.0f * p.z, 1.0f);
}

// ---------------------------------------------------------------------------
// Kernel 3: WMMA distance GEMM (d2' = |p|^2 - 2 q.p) fused with per-lane
// running top-4 selection.
//  - TDM stages the whole batch point slab (128KB) into LDS once per block;
//    all 8 waves then run the 512-iteration M loop purely out of LDS.
//  - Insert network is branchless (cndmask); a wave-uniform __any() early-out
//    skips it for tiles that cannot improve any lane's top-4 (EXEC stays all-1s
//    around every v_wmma, per ISA requirement).
// ---------------------------------------------------------------------------
__global__ __launch_bounds__(256) void knn_wmma(const float4* __restrict__ P,
                                                const float4* __restrict__ Q,
                                                int* __restrict__ nn_idx,
                                                float* __restrict__ nn_d2) {
    extern __shared__ float ldsP[];      // NPTS * 4 floats = 128KB

    int wave = (blockIdx.x * blockDim.x + threadIdx.x) >> 5;   // 0..1023
    int lane = threadIdx.x & 31;
    int b  = blockIdx.x >> 5;           // 8 waves/block, 256 query-groups/batch
    int q0 = (wave & 255) << 4;         // first of 16 queries
    int col = lane & 15;
    int hi  = (lane >= 16) ? 1 : 0;

    // ---- stage P[b] into LDS via the Tensor Data Mover (wave 0 issues) ----
#if __has_builtin(__builtin_amdgcn_tensor_load_to_lds)
    if ((threadIdx.x >> 5) == 0) {
        unsigned long long ga = (unsigned long long)(const void*)(P + (size_t)b * NPTS);
        u32x4 g0;
        g0[0] = 1u;                                            // count=1 (valid), user mode
        g0[1] = 0u;                                            // lds_addr = 0 (dynamic LDS base)
        g0[2] = (unsigned)(ga & 0xFFFFFFFFull);                // global_addr[31:0]
        g0[3] = (unsigned)((ga >> 32) & 0x01FFFFFFull)         // global_addr[56:32]
              | (2u << 30);                                    // type = 2 ("image")
        i32x8 g1;
        g1[0] = (int)(3u << 16);          // data_size = 3 (8 bytes), no mask/flags
        g1[1] = (int)((NPTS * 2) << 16);  // tensor_dim0 lo16 (=16384 8B units) in [31:16]
        g1[2] = (int)(1u << 16);          // tensor_dim0 hi16 = 0 ; tensor_dim1 = 1
        g1[3] = (int)((NPTS * 2) << 16);  // tensor_dim1 hi = 0 ; tile_dim0 = 16384
        g1[4] = 1;                        // tile_dim1 = 1 ; tile_dim2 = 0
        g1[5] = NPTS * 2;                 // tensor_dim0_stride lo32
        g1[6] = 0;                        // stride0 hi ; tensor_dim1_stride lo
        g1[7] = 0;                        // tensor_dim1_stride hi
        i32x4 gz4 = {};
        i32x8 gz8 = {};
        __builtin_amdgcn_tensor_load_to_lds(g0, g1, gz4, gz4, gz8, 0);
        __builtin_amdgcn_s_wait_tensorcnt(0);
    }
#else
    {
        const float* src = (const float*)(P + (size_t)b * NPTS);
        for (int i = threadIdx.x; i < NPTS * 4; i += 256) ldsP[i] = src[i];
    }
#endif
    __syncthreads();

    // ---- B operand (constant over M loop): K-pair (0,1) lo half, (2,3) hi ----
    const float* qp = (const float*)(Q + (b * NPIX + q0 + col));
    __builtin_prefetch(qp, 0, 0);
    v2f Bv;
    Bv.x = qp[hi ? 2 : 0];
    Bv.y = qp[hi ? 3 : 1];

    float bd0 = INFINITY, bd1 = INFINITY, bd2 = INFINITY, bd3 = INFINITY;
    int   bi0 = 0, bi1 = 0, bi2 = 0, bi3 = 0;

    // Branchless sorted insert (ascending); strict < keeps earlier index on ties
    auto ins = [&](float v, int id) {
        bool c0 = v < bd0, c1 = v < bd1, c2 = v < bd2, c3 = v < bd3;
        float nb0 = c0 ? v   : bd0;
        float nb1 = c0 ? bd0 : (c1 ? v : bd1);
        float nb2 = c1 ? bd1 : (c2 ? v : bd2);
        float nb3 = c2 ? bd2 : (c3 ? v : bd3);
        int   ni0 = c0 ? id  : bi0;
        int   ni1 = c0 ? bi0 : (c1 ? id : bi1);
        int   ni2 = c1 ? bi1 : (c2 ? id : bi2);
        int   ni3 = c2 ? bi2 : (c3 ? id : bi3);
        bd0 = nb0; bd1 = nb1; bd2 = nb2; bd3 = nb3;
        bi0 = ni0; bi1 = ni1; bi2 = ni2; bi3 = ni3;
    };

#pragma unroll 2
    for (int m0 = 0; m0 < NPTS; m0 += 16) {
        // A operand from LDS: point (m0+col), K-pair (x,y) lo half, (z,|p|^2) hi
        float2 a2 = *(const float2*)(ldsP + (size_t)(m0 + col) * 4 + (hi ? 2 : 0));
        v2f Av;
        Av.x = a2.x;
        Av.y = a2.y;
        v8f Cz = {};
        v8f Dv = __builtin_amdgcn_wmma_f32_16x16x4_f32(
            /*neg_a=*/false, Av, /*neg_b=*/false, Bv,
            /*c_mod=*/(short)0, Cz, /*reuse_a=*/false, /*reuse_b=*/false);

        float m8 = fminf(fminf(fminf(Dv[0], Dv[1]), fminf(Dv[2], Dv[3])),
                         fminf(fminf(Dv[4], Dv[5]), fminf(Dv[6], Dv[7])));
        if (__any(m8 < bd3)) {   // wave-uniform branch; EXEC untouched
#pragma unroll
            for (int r = 0; r < 8; ++r) {
                ins(Dv[r], m0 + r + hi * 8);
            }
        }
    }

    // Merge the two half-wave lists for each column (snapshot before insert)
    float p0 = __shfl_xor(bd0, 16, 32);
    float p1 = __shfl_xor(bd1, 16, 32);
    float p2 = __shfl_xor(bd2, 16, 32);
    float p3 = __shfl_xor(bd3, 16, 32);
    int   j0 = __shfl_xor(bi0, 16, 32);
    int   j1 = __shfl_xor(bi1, 16, 32);
    int   j2 = __shfl_xor(bi2, 16, 32);
    int   j3 = __shfl_xor(bi3, 16, 32);
    ins(p0, j0); ins(p1, j1); ins(p2, j2); ins(p3, j3);

    if (!hi) {
        int qi = b * NPIX + q0 + col;
        nn_idx[qi * 4 + 0] = bi0;  nn_d2[qi * 4 + 0] = bd0;
        nn_idx[qi * 4 + 1] = bi1;  nn_d2[qi * 4 + 1] = bd1;
        nn_idx[qi * 4 + 2] = bi2;  nn_d2[qi * 4 + 2] = bd2;
        nn_idx[qi * 4 + 3] = bi3;  nn_d2[qi * 4 + 3] = bd3;
    }
}

// ---------------------------------------------------------------------------
// Kernel 4: gather NN, softmax weights, blend pts/feats, project, z-buffer min
// One block (64 threads = channels) per query.
// softmax(10*negd2 + log(conf)) == softmax(-10*d2' + log(conf))  (row-const inv.)
// ---------------------------------------------------------------------------
__global__ __launch_bounds__(64) void gather_project(
        const int* __restrict__ nn_idx, const float* __restrict__ nn_d2,
        const float4* __restrict__ P,
        const float* __restrict__ dc1, const float* __restrict__ dc2,
        const float* __restrict__ fea1, const float* __restrict__ fea2,
        const float* __restrict__ Kc, const float* __restrict__ ORT,
        float* __restrict__ ffeat, int* __restrict__ sp_idx,
        float* __restrict__ sp_z, unsigned* __restrict__ zbuf) {
    int qi = blockIdx.x;
    int b  = qi >> 12;
    __shared__ float sw[4];
    __shared__ int   snn[4];

    if (threadIdx.x == 0) {
        float dd[4], cf[4], s[4], e[4], w[4];
        int   id[4];
#pragma unroll
        for (int k = 0; k < 4; ++k) {
            id[k] = nn_idx[qi * 4 + k];
            dd[k] = nn_d2[qi * 4 + k];
            int j = id[k];
            cf[k] = (j < NPIX) ? dc1[b * NPIX + j] : dc2[b * NPIX + j - NPIX];
        }
        float mx = -INFINITY;
#pragma unroll
        for (int k = 0; k < 4; ++k) { s[k] = -10.0f * dd[k] + logf(cf[k] + 1e-6f); mx = fmaxf(mx, s[k]); }
        float sum = 0.0f;
#pragma unroll
        for (int k = 0; k < 4; ++k) { e[k] = expf(s[k] - mx); sum += e[k]; }
        float inv = 1.0f / sum;
        float fp0 = 0.0f, fp1 = 0.0f, fp2 = 0.0f;
#pragma unroll
        for (int k = 0; k < 4; ++k) {
            w[k] = e[k] * inv;
            float4 p = P[b * NPTS + id[k]];
            fp0 += w[k] * p.x; fp1 += w[k] * p.y; fp2 += w[k] * p.z;
            sw[k] = w[k]; snn[k] = id[k];
        }
        // xyp = K @ (output_RT @ [fp,1])
        const float* R  = ORT + b * 16;
        const float* Km = Kc  + b * 16;
        float t0 = R[0]*fp0 + R[1]*fp1 + R[2] *fp2 + R[3];
        float t1 = R[4]*fp0 + R[5]*fp1 + R[6] *fp2 + R[7];
        float t2 = R[8]*fp0 + R[9]*fp1 + R[10]*fp2 + R[11];
        float t3 = R[12]*fp0 + R[13]*fp1 + R[14]*fp2 + R[15];
        float x0 = Km[0]*t0 + Km[1]*t1 + Km[2] *t2 + Km[3] *t3;
        float x1 = Km[4]*t0 + Km[5]*t1 + Km[6] *t2 + Km[7] *t3;
        float x2 = Km[8]*t0 + Km[9]*t1 + Km[10]*t2 + Km[11]*t3;
        float z  = x2;
        bool  msk = fabsf(z) < EPSV;
        float zs  = msk ? EPSV : z;
        float sx  = msk ? 10.0f : -(x0 * 2.0f / zs / (float)(WRES - 1) - 1.0f);
        float sy  = msk ? 10.0f : -(x1 * 2.0f / zs / (float)(WRES - 1) - 1.0f);
        float sz  = msk ? 10.0f : z;
        int u = (int)rintf((sx + 1.0f) * 0.5f * (float)(WRES - 1));
        int v = (int)rintf((sy + 1.0f) * 0.5f * (float)(WRES - 1));
        bool valid = (u >= 0) && (u < WRES) && (v >= 0) && (v < WRES) &&
                     (sz > EPSV * 0.5f) && (sx < 9.0f);
        int oidx = valid ? (b * NPIX + v * WRES + u) : -1;
        sp_idx[qi] = oidx;
        sp_z[qi]   = sz;
        if (valid) atomicMin(&zbuf[oidx], __float_as_uint(sz));  // sz > 0 here
    }
    __syncthreads();

    int c = threadIdx.x;
    float acc = 0.0f;
#pragma unroll
    for (int k = 0; k < 4; ++k) {
        int j = snn[k];
        float fv = (j < NPIX) ? fea1[(b * CCH + c) * NPIX + j]
                              : fea2[(b * CCH + c) * NPIX + j - NPIX];
        acc += sw[k] * fv;
    }
    ffeat[(size_t)qi * CCH + c] = acc;
}

// ---------------------------------------------------------------------------
// Kernel 5: z-buffer winners scatter-add features + depth into d_out
// d_out layout: res[:, :3] | res[:, 3:] | depth   (tuple flattened)
// ---------------------------------------------------------------------------
__global__ __launch_bounds__(64) void splat_add(const int* __restrict__ sp_idx,
                                                const float* __restrict__ sp_z,
                                                const unsigned* __restrict__ zbuf,
                                                const float* __restrict__ ffeat,
                                                float* __restrict__ out) {
    int qi   = blockIdx.x;
    int oidx = sp_idx[qi];
    if (oidx < 0) return;
    float z = sp_z[qi];
    if (__float_as_uint(z) != zbuf[oidx]) return;   // only z-buffer winners (ties all add)
    int b   = qi >> 12;
    int pix = oidx & (NPIX - 1);
    int c   = threadIdx.x;
    float f = ffeat[(size_t)qi * CCH + c];
    if (c < 3)
        atomicAdd(out + ((size_t)b * 3 + c) * NPIX + pix, f);
    else
        atomicAdd(out + (size_t)BATCH * 3 * NPIX + ((size_t)b * 61 + (c - 3)) * NPIX + pix, f);
    if (c == 0)
        atomicAdd(out + (size_t)BATCH * CCH * NPIX + (size_t)b * NPIX + pix, z);
}

// ---------------------------------------------------------------------------
extern "C" void kernel_launch(void* const* d_in, const int* in_sizes, int n_in,
                              void* d_out, int out_size, void* d_ws, size_t ws_size,
                              hipStream_t stream) {
    const float* fea1 = (const float*)d_in[0];
    const float* fea2 = (const float*)d_in[1];
    const float* pp1  = (const float*)d_in[2];
    const float* pp2  = (const float*)d_in[3];
    const float* Kc   = (const float*)d_in[4];
    const float* Kinv = (const float*)d_in[5];
    const float* Ri1  = (const float*)d_in[8];
    const float* Ri2  = (const float*)d_in[9];
    const float* ORT  = (const float*)d_in[10];
    const float* dc1  = (const float*)d_in[12];
    const float* dc2  = (const float*)d_in[13];
    float* out = (float*)d_out;

    char* ws = (char*)d_ws;
    float4*   P     = (float4*)ws;   ws += (size_t)BATCH * NPTS * sizeof(float4);
    float4*   Q     = (float4*)ws;   ws += (size_t)BATCH * NPIX * sizeof(float4);
    int*      nn_i  = (int*)ws;      ws += (size_t)BATCH * NPIX * KNN * sizeof(int);
    float*    nn_d  = (float*)ws;    ws += (size_t)BATCH * NPIX * KNN * sizeof(float);
    float*    ffeat = (float*)ws;    ws += (size_t)BATCH * NPIX * CCH * sizeof(float);
    int*      sp_i  = (int*)ws;      ws += (size_t)BATCH * NPIX * sizeof(int);
    float*    sp_z  = (float*)ws;    ws += (size_t)BATCH * NPIX * sizeof(float);
    unsigned* zbuf  = (unsigned*)ws;

    init_kernel<<<(out_size + 255) / 256, 256, 0, stream>>>(out, out_size, zbuf);
    prep_points<<<(BATCH * NPTS + 255) / 256, 256, 0, stream>>>(pp1, pp2, Kinv, Ri1, Ri2, P);
    prep_queries<<<(BATCH * NPIX + 255) / 256, 256, 0, stream>>>(P, Q);
    // 1024 waves (16 queries each) = 128 blocks of 256; 128KB dynamic LDS for P slab
    knn_wmma<<<(BATCH * NPIX / 16) * 32 / 256, 256, NPTS * 4 * sizeof(float), stream>>>(
        P, Q, nn_i, nn_d);
    gather_project<<<BATCH * NPIX, 64, 0, stream>>>(nn_i, nn_d, P, dc1, dc2, fea1, fea2,
                                                    Kc, ORT, ffeat, sp_i, sp_z, zbuf);
    splat_add<<<BATCH * NPIX, 64, 0, stream>>>(sp_i, sp_z, zbuf, ffeat, out);
}